// GAT_35270271435478
// MI455X (gfx1250) — compile-verified
//
#include <hip/hip_runtime.h>

typedef __attribute__((ext_vector_type(2))) float v2f;
typedef __attribute__((ext_vector_type(8))) float v8f;

#define NEG_SLOPE 0.2f

__device__ __forceinline__ float lrelu(float x) { return x > 0.0f ? x : NEG_SLOPE * x; }

// Monotone order-preserving float->uint encoding for atomicMax-based segment max.
__device__ __forceinline__ unsigned enc_f32(float f) {
    unsigned u = __float_as_uint(f);
    return (u & 0x80000000u) ? ~u : (u | 0x80000000u);
}
__device__ __forceinline__ float dec_f32(unsigned e) {
    unsigned u = (e & 0x80000000u) ? (e & 0x7FFFFFFFu) : ~e;
    return __uint_as_float(u);
}
// Guaranteed-hardware fp32 atomic add (global_atomic_add_f32), never a CAS loop.
__device__ __forceinline__ void atomic_add_f32(float* p, float v) {
    unsafeAtomicAdd(p, v);
}
__device__ __forceinline__ void wait_asynccnt0() {
#if __has_builtin(__builtin_amdgcn_s_wait_asynccnt)
    __builtin_amdgcn_s_wait_asynccnt(0);
#else
    asm volatile("s_wait_asynccnt 0" ::: "memory");
#endif
}

__global__ void fill_u32_kernel(unsigned* __restrict__ p, unsigned v, int n) {
    int i = blockIdx.x * blockDim.x + threadIdx.x;
    if (i < n) p[i] = v;
}

// ---------------------------------------------------------------------------
// Out[M x NCOLS] = A[M x K] @ W[K x NCOLS], exact fp32 via V_WMMA_F32_16X16X4_F32.
// One block = 16 output rows; wave w owns columns [16w, 16w+16).
// A tile DMA'd to LDS with global_load_async_to_lds_b128 (ASYNCcnt-tracked),
// row pad +4 floats to spread LDS banks; W streamed from L2 (128 KB, hot).
// ---------------------------------------------------------------------------
template <int K, int NW>
__global__ __launch_bounds__(NW * 32) void wmma_gemm_kernel(
    const float* __restrict__ A, const float* __restrict__ W,
    float* __restrict__ Out, int M) {
    constexpr int NCOLS = NW * 16;
    constexpr int LDSROW = K + 4;
    __shared__ float tileA[16 * LDSROW];

    const int tid  = threadIdx.x;
    const int row0 = blockIdx.x * 16;

    // Cooperative async DMA of the 16 x K A-tile into LDS, 16 bytes per issue.
    for (int i = tid; i < (16 * K) / 4; i += NW * 32) {
        int g = i * 4;
        int r = g / K, c = g % K;
        unsigned lds_off = (unsigned)(size_t)(const void*)(tileA + r * LDSROW + c);
        unsigned goff    = (unsigned)(((row0 + r) * K + c) * 4);
        asm volatile("global_load_async_to_lds_b128 %0, %1, %2"
                     :: "v"(lds_off), "v"(goff), "s"(A)
                     : "memory");
    }
    wait_asynccnt0();      // my wave's DMA complete
    __syncthreads();       // all waves' tiles visible

    const int wave = tid >> 5;
    const int lane = tid & 31;
    const int m    = lane & 15;   // A row / (B,C,D column)
    const int kh   = lane >> 4;   // K-half for A/B frags, M-half for C/D
    const int col  = wave * 16 + m;

    v8f acc = {};
    for (int k0 = 0; k0 < K; k0 += 4) {
        v2f a, b;
        // A 16x4 frag: lanes 0-15 -> K=k0,k0+1 ; lanes 16-31 -> K=k0+2,k0+3
        a.x = tileA[m * LDSROW + k0 + 2 * kh + 0];
        a.y = tileA[m * LDSROW + k0 + 2 * kh + 1];
        // B 4x16 frag: same K-half split across lane halves, column = lane&15
        b.x = W[(size_t)(k0 + 2 * kh + 0) * NCOLS + col];
        b.y = W[(size_t)(k0 + 2 * kh + 1) * NCOLS + col];
        acc = __builtin_amdgcn_wmma_f32_16x16x4_f32(
            false, a, false, b, (short)0, acc, false, false);
    }

    // D layout: vgpr r -> row (r + 8*kh), column = lane&15.
#pragma unroll
    for (int r = 0; r < 8; ++r)
        Out[(size_t)(row0 + r + 8 * kh) * NCOLS + col] = acc[r];
}

// ---------------------------------------------------------------------------
// Per-node attention logits: as_n[n,h] = sum_c h[n,h,c]*a_s[h,c]  (same for a_d).
// One wave32 per node; lane l covers channels {l, l+32, ...}; butterfly reduce.
// ---------------------------------------------------------------------------
template <int H, int C>
__global__ void attn_dots_kernel(const float* __restrict__ hfeat,
                                 const float* __restrict__ a_s,
                                 const float* __restrict__ a_d,
                                 float* __restrict__ as_n, float* __restrict__ ad_n,
                                 int N) {
    constexpr int HC = H * C;
    const int wave = threadIdx.x >> 5, lane = threadIdx.x & 31;
    const int node = blockIdx.x * (blockDim.x >> 5) + wave;
    if (node >= N) return;

    float ps[H], pd[H];
#pragma unroll
    for (int h = 0; h < H; ++h) { ps[h] = 0.0f; pd[h] = 0.0f; }

#pragma unroll
    for (int j = 0; j < HC / 32; ++j) {
        const int c  = 32 * j + lane;
        const int hh = (32 * j) / C;           // uniform per chunk (C % 32 == 0)
        const int ai = (32 * j) % C + lane;    // channel within head
        float v = hfeat[(size_t)node * HC + c];
        ps[hh] += v * a_s[hh * C + ai];
        pd[hh] += v * a_d[hh * C + ai];
    }
#pragma unroll
    for (int h = 0; h < H; ++h) {
        float s = ps[h], d = pd[h];
        for (int off = 16; off; off >>= 1) {
            s += __shfl_xor(s, off);
            d += __shfl_xor(d, off);
        }
        if (lane == 0) {
            as_n[node * H + h] = s;
            ad_n[node * H + h] = d;
        }
    }
}

// Pass 1: segment max over incoming edges (atomicMax on monotone-encoded floats)
template <int H>
__global__ void edge_max_kernel(const int* __restrict__ src, const int* __restrict__ dst,
                                const float* __restrict__ as_n, const float* __restrict__ ad_n,
                                unsigned* __restrict__ menc, int E) {
    int idx = blockIdx.x * blockDim.x + threadIdx.x;
    if (idx >= E * H) return;
    int e = idx / H, h = idx - e * H;
    int s = src[e], d = dst[e];
    float ev = lrelu(as_n[s * H + h] + ad_n[d * H + h]);
    atomicMax(&menc[d * H + h], enc_f32(ev));
}

// Pass 2: ex = exp(e - max); accumulate softmax denominator per dst node
template <int H>
__global__ void edge_exp_kernel(const int* __restrict__ src, const int* __restrict__ dst,
                                const float* __restrict__ as_n, const float* __restrict__ ad_n,
                                const unsigned* __restrict__ menc,
                                float* __restrict__ ex, float* __restrict__ den, int E) {
    int idx = blockIdx.x * blockDim.x + threadIdx.x;
    if (idx >= E * H) return;
    int e = idx / H, h = idx - e * H;
    int s = src[e], d = dst[e];
    float ev = lrelu(as_n[s * H + h] + ad_n[d * H + h]);
    float xv = __expf(ev - dec_f32(menc[d * H + h]));
    ex[(size_t)e * H + h] = xv;
    atomic_add_f32(&den[d * H + h], xv);
}

// Pass 3: out[dst] += h[src] * alpha. One wave32 per edge, lanes span channels.
template <int H, int C>
__global__ void edge_scatter_kernel(const int* __restrict__ src, const int* __restrict__ dst,
                                    const float* __restrict__ hfeat,
                                    const float* __restrict__ ex, const float* __restrict__ den,
                                    float* __restrict__ out, int E) {
    constexpr int HC = H * C;
    const int wave = threadIdx.x >> 5, lane = threadIdx.x & 31;
    const int e = blockIdx.x * (blockDim.x >> 5) + wave;
    if (e >= E) return;
    const int s = src[e], d = dst[e];
#pragma unroll
    for (int j = 0; j < HC / 32; ++j) {
        const int c  = 32 * j + lane;
        const int hh = (32 * j) / C;
        float alpha = ex[(size_t)e * H + hh] / (den[d * H + hh] + 1e-16f);
        float v = hfeat[(size_t)s * HC + c] * alpha;
        atomic_add_f32(&out[(size_t)d * HC + c], v);
    }
}

// Epilogue: io[i] = act(io[i] + b[i % HC])
__global__ void bias_act_kernel(float* __restrict__ io, const float* __restrict__ b,
                                int HC, int n, int do_relu) {
    int i = blockIdx.x * blockDim.x + threadIdx.x;
    if (i >= n) return;
    float v = io[i] + b[i % HC];
    io[i] = do_relu ? fmaxf(v, 0.0f) : v;
}

extern "C" void kernel_launch(void* const* d_in, const int* in_sizes, int n_in,
                              void* d_out, int out_size, void* d_ws, size_t ws_size,
                              hipStream_t stream) {
    (void)in_sizes; (void)n_in; (void)out_size; (void)ws_size;
    constexpr int N = 50000, E = 800000;
    constexpr int H1 = 4, C1 = 32, HC1 = 128, HC2 = 64;

    const float* x    = (const float*)d_in[0];
    const int*   ei   = (const int*)d_in[1];
    const int*   src  = ei;
    const int*   dst  = ei + E;
    const float* W1   = (const float*)d_in[2];
    const float* a_s1 = (const float*)d_in[3];
    const float* a_d1 = (const float*)d_in[4];
    const float* b1   = (const float*)d_in[5];
    const float* W2   = (const float*)d_in[6];
    const float* a_s2 = (const float*)d_in[7];
    const float* a_d2 = (const float*)d_in[8];
    const float* b2   = (const float*)d_in[9];
    float* out = (float*)d_out;

    char* wsp = (char*)d_ws;
    auto alloc = [&](size_t bytes) -> void* {
        void* p = wsp;
        wsp += (bytes + 255) & ~(size_t)255;
        return p;
    };
    float*    h1   = (float*)alloc((size_t)N * HC1 * 4);
    float*    as1  = (float*)alloc((size_t)N * H1 * 4);
    float*    ad1  = (float*)alloc((size_t)N * H1 * 4);
    unsigned* m1   = (unsigned*)alloc((size_t)N * H1 * 4);
    float*    den1 = (float*)alloc((size_t)N * H1 * 4);
    float*    ex1  = (float*)alloc((size_t)E * H1 * 4);
    float*    o1   = (float*)alloc((size_t)N * HC1 * 4);
    float*    h2   = (float*)alloc((size_t)N * HC2 * 4);
    float*    as2  = (float*)alloc((size_t)N * 4);
    float*    ad2  = (float*)alloc((size_t)N * 4);
    unsigned* m2   = (unsigned*)alloc((size_t)N * 4);
    float*    den2 = (float*)alloc((size_t)N * 4);
    float*    ex2  = (float*)alloc((size_t)E * 4);

    auto grid1 = [](int n) { return (n + 255) / 256; };

    // ---- init accumulators (max-enc buffers to 0 == below enc(-inf), sums to 0)
    fill_u32_kernel<<<grid1(N * H1), 256, 0, stream>>>(m1, 0u, N * H1);
    fill_u32_kernel<<<grid1(N * H1), 256, 0, stream>>>((unsigned*)den1, 0u, N * H1);
    fill_u32_kernel<<<grid1(N * HC1), 256, 0, stream>>>((unsigned*)o1, 0u, N * HC1);
    fill_u32_kernel<<<grid1(N), 256, 0, stream>>>(m2, 0u, N);
    fill_u32_kernel<<<grid1(N), 256, 0, stream>>>((unsigned*)den2, 0u, N);
    fill_u32_kernel<<<grid1(N * HC2), 256, 0, stream>>>((unsigned*)out, 0u, N * HC2);

    // ---- layer 1 -----------------------------------------------------------
    wmma_gemm_kernel<256, 8><<<N / 16, 256, 0, stream>>>(x, W1, h1, N);
    attn_dots_kernel<H1, C1><<<(N + 7) / 8, 256, 0, stream>>>(h1, a_s1, a_d1, as1, ad1, N);
    edge_max_kernel<H1><<<grid1(E * H1), 256, 0, stream>>>(src, dst, as1, ad1, m1, E);
    edge_exp_kernel<H1><<<grid1(E * H1), 256, 0, stream>>>(src, dst, as1, ad1, m1, ex1, den1, E);
    edge_scatter_kernel<H1, C1><<<(E + 7) / 8, 256, 0, stream>>>(src, dst, h1, ex1, den1, o1, E);
    bias_act_kernel<<<grid1(N * HC1), 256, 0, stream>>>(o1, b1, HC1, N * HC1, 1);

    // ---- layer 2 -----------------------------------------------------------
    wmma_gemm_kernel<128, 4><<<N / 16, 128, 0, stream>>>(o1, W2, h2, N);
    attn_dots_kernel<1, 64><<<(N + 7) / 8, 256, 0, stream>>>(h2, a_s2, a_d2, as2, ad2, N);
    edge_max_kernel<1><<<grid1(E), 256, 0, stream>>>(src, dst, as2, ad2, m2, E);
    edge_exp_kernel<1><<<grid1(E), 256, 0, stream>>>(src, dst, as2, ad2, m2, ex2, den2, E);
    edge_scatter_kernel<1, 64><<<(E + 7) / 8, 256, 0, stream>>>(src, dst, h2, ex2, den2, out, E);
    bias_act_kernel<<<grid1(N * HC2), 256, 0, stream>>>(out, b2, HC2, N * HC2, 0);
}